// M1V1V2Perception_48146583388475
// MI455X (gfx1250) — compile-verified
//
#include <hip/hip_runtime.h>
#include <math.h>

// Problem constants (match reference)
#define B_    16
#define F_    6
#define H_    512
#define W_    512
#define N_    (H_*W_)      // 262144 pixels per batch
#define K_    8
#define ITERS_ 12
#define TEMP_  0.15f
#define EPS_   1e-6f
#define NBLK   64          // blocks per batch for the assign pass

typedef float v2f __attribute__((ext_vector_type(2)));
typedef float v8f __attribute__((ext_vector_type(8)));

#if defined(__gfx1250__) && __has_builtin(__builtin_amdgcn_wmma_f32_16x16x4_f32)
#define HAVE_WMMA_F32 1
#else
#define HAVE_WMMA_F32 0
#endif

// xor-16 half-wave swap. Prefer v_permlanex16_b32 (VALU pipe, identity lane
// selects => lane i reads lane i^16) over ds_bpermute (LDS pipe + dscnt wait).
__device__ __forceinline__ float permx16(float v) {
#if defined(__gfx1250__) && __has_builtin(__builtin_amdgcn_permlanex16)
  int iv = __builtin_bit_cast(int, v);
  int r  = __builtin_amdgcn_permlanex16(iv, iv, 0x76543210, 0xfedcba98,
                                        false, false);
  return __builtin_bit_cast(float, r);
#else
  return __shfl_xor(v, 16, 32);
#endif
}

// ---------------------------------------------------------------------------
// Pass 1: per-(b,f) mean and 1/clip(std,eps), unbiased std (ddof=1).
// ---------------------------------------------------------------------------
__global__ void stats_kernel(const float* __restrict__ feat,
                             float* __restrict__ meanv,
                             float* __restrict__ rstdv) {
  const int bf = blockIdx.x;                        // 0 .. B*F-1
  const float4* p4 = (const float4*)(feat + (size_t)bf * N_);
  float s = 0.f, s2 = 0.f;
  for (int i = threadIdx.x; i < N_/4; i += blockDim.x) {
    float4 v = p4[i];
    s  += v.x + v.y + v.z + v.w;
    s2 += v.x*v.x + v.y*v.y + v.z*v.z + v.w*v.w;
  }
  #pragma unroll
  for (int o = 16; o > 0; o >>= 1) {
    s  += __shfl_xor(s,  o, 32);
    s2 += __shfl_xor(s2, o, 32);
  }
  __shared__ float ls[2][8];
  const int wid = threadIdx.x >> 5, lid = threadIdx.x & 31;
  if (lid == 0) { ls[0][wid] = s; ls[1][wid] = s2; }
  __syncthreads();
  if (threadIdx.x == 0) {
    float S = 0.f, S2 = 0.f;
    for (int w = 0; w < 8; ++w) { S += ls[0][w]; S2 += ls[1][w]; }
    float mean = S / (float)N_;
    float var  = (S2 - S * mean) / (float)(N_ - 1);   // unbiased (ddof=1)
    float sd   = sqrtf(fmaxf(var, 0.f));
    meanv[bf] = mean;
    rstdv[bf] = 1.f / fmaxf(sd, EPS_);
  }
}

// ---------------------------------------------------------------------------
// Pass 2: gather standardized initial centroids C0[b][k][f].
// ---------------------------------------------------------------------------
__global__ void init_centroids(const float* __restrict__ feat,
                               const int*   __restrict__ idx,
                               const float* __restrict__ meanv,
                               const float* __restrict__ rstdv,
                               float* __restrict__ C) {
  const int t = threadIdx.x;
  if (t >= B_ * K_) return;
  const int b = t / K_, k = t % K_;
  const int j = idx[b*K_ + k];
  #pragma unroll
  for (int f = 0; f < F_; ++f) {
    const int bf = b*F_ + f;
    float v = feat[(size_t)bf * N_ + j];
    C[(b*K_ + k)*F_ + f] = (v - meanv[bf]) * rstdv[bf];
  }
}

// ---------------------------------------------------------------------------
// Assignment pass. FINAL=false: accumulate per-block {sum S_k, sum S_k*x_f}.
// FINAL=true: write normalized S to out (B,K,H,W) with NT stores.
//
// xc = X . C^T with a single V_WMMA_F32_16X16X4_F32 accumulator chain:
//   tile1 (pixels of lanes 0..15):  clusters in A rows 0..7  -> D lanes 0..15
//   tile2 (pixels of lanes 16..31): clusters in A rows 8..15 -> D lanes 16..31
// Row sets are disjoint, so 4 chained WMMAs leave xc[k] in-place for every
// lane's own pixel -- no post-merge cross-lane moves at all.
// ---------------------------------------------------------------------------
template<bool FINAL>
__global__ void assign_kernel(const float* __restrict__ feat,
                              const float* __restrict__ meanv,
                              const float* __restrict__ rstdv,
                              const float* __restrict__ C,
                              float* __restrict__ partials,
                              float* __restrict__ out) {
  const int b    = blockIdx.y;
  const int tid  = threadIdx.x;
  const int lid  = tid & 31;
  const int wid  = tid >> 5;
  const int cl   = lid & 15;     // A-matrix row this lane supplies
  const int half = lid >> 4;     // 0: K-slots {0,1}; 1: K-slots {2,3}
  const bool lo  = (half == 0);

  const float* Cb = C + b*K_*F_;

  float mu[F_], rs[F_];
  #pragma unroll
  for (int f = 0; f < F_; ++f) { mu[f] = meanv[b*F_+f]; rs[f] = rstdv[b*F_+f]; }

  // ||c_k||^2 (wave-uniform, cached loads)
  float c2[K_];
  #pragma unroll
  for (int k = 0; k < K_; ++k) {
    float s = 0.f;
    #pragma unroll
    for (int f = 0; f < F_; ++f) { float c = Cb[k*F_+f]; s += c*c; }
    c2[k] = s;
  }

#if HAVE_WMMA_F32
  // A registers. lo set: clusters in rows 0..7 (tile1).
  //              hi set: clusters in rows 8..15 (tile2).
  v2f a0lo = {0.f,0.f}, a1lo = {0.f,0.f};
  v2f a0hi = {0.f,0.f}, a1hi = {0.f,0.f};
  if (cl < K_) {
    a0lo.x = Cb[cl*F_ + 2*half];
    a0lo.y = Cb[cl*F_ + 2*half + 1];
    if (lo) { a1lo.x = Cb[cl*F_ + 4]; a1lo.y = Cb[cl*F_ + 5]; }
  } else {
    const int ck = cl - K_;
    a0hi.x = Cb[ck*F_ + 2*half];
    a0hi.y = Cb[ck*F_ + 2*half + 1];
    if (lo) { a1hi.x = Cb[ck*F_ + 4]; a1hi.y = Cb[ck*F_ + 5]; }
  }
#else
  float CL[K_][F_];
  #pragma unroll
  for (int k = 0; k < K_; ++k)
    #pragma unroll
    for (int f = 0; f < F_; ++f) CL[k][f] = Cb[k*F_+f];
#endif

  float dacc[K_];
  float cacc[K_][F_];
  if (!FINAL) {
    #pragma unroll
    for (int k = 0; k < K_; ++k) {
      dacc[k] = 0.f;
      #pragma unroll
      for (int f = 0; f < F_; ++f) cacc[k][f] = 0.f;
    }
  }

  const float* fb = feat + (size_t)b * F_ * N_;
  const int pixPerBlock = N_ / NBLK;                 // 4096
  const int base = blockIdx.x * pixPerBlock;

  for (int p0 = base + wid*32; p0 < base + pixPerBlock; p0 += blockDim.x) {
    const int pix = p0 + lid;                        // this lane's own pixel
    float x[F_];
    #pragma unroll
    for (int f = 0; f < F_; ++f)
      x[f] = (fb[(size_t)f*N_ + pix] - mu[f]) * rs[f];

    float xc[K_];
#if HAVE_WMMA_F32
    float sx[F_];
    #pragma unroll
    for (int f = 0; f < F_; ++f) sx[f] = permx16(x[f]);   // VALU half swap
    // B tiles (4 features x 16 pixels) in split VGPR layout:
    v2f b10 = { lo ? x[0]  : sx[2], lo ? x[1]  : sx[3] }; // tile1 f0..3
    v2f b11 = { lo ? x[4]  : 0.f,   lo ? x[5]  : 0.f   }; // tile1 f4..7
    v2f b20 = { lo ? sx[0] : x[2],  lo ? sx[1] : x[3]  }; // tile2 f0..3
    v2f b21 = { lo ? sx[4] : 0.f,   lo ? sx[5] : 0.f   }; // tile2 f4..7
    v8f acc = {0.f,0.f,0.f,0.f,0.f,0.f,0.f,0.f};
    acc = __builtin_amdgcn_wmma_f32_16x16x4_f32(false, a1hi, false, b21, (short)0, acc, false, false);
    acc = __builtin_amdgcn_wmma_f32_16x16x4_f32(false, a0hi, false, b20, (short)0, acc, false, false);
    acc = __builtin_amdgcn_wmma_f32_16x16x4_f32(false, a1lo, false, b11, (short)0, acc, false, false);
    acc = __builtin_amdgcn_wmma_f32_16x16x4_f32(false, a0lo, false, b10, (short)0, acc, false, false);
    #pragma unroll
    for (int k = 0; k < K_; ++k) xc[k] = acc[k];          // in-place result
#else
    #pragma unroll
    for (int k = 0; k < K_; ++k) {
      float s = 0.f;
      #pragma unroll
      for (int f = 0; f < F_; ++f) s += CL[k][f] * x[f];
      xc[k] = s;
    }
#endif

    float x2 = 0.f;
    #pragma unroll
    for (int f = 0; f < F_; ++f) x2 += x[f]*x[f];

    // softmax over K of -clip(d2,0)/TEMP
    float lg[K_], mx = -3.4e38f;
    #pragma unroll
    for (int k = 0; k < K_; ++k) {
      float d2 = fmaxf(x2 + c2[k] - 2.f*xc[k], 0.f);
      lg[k] = d2 * (-1.f/TEMP_);
      mx = fmaxf(mx, lg[k]);
    }
    float e[K_], ssum = 0.f;
    #pragma unroll
    for (int k = 0; k < K_; ++k) { e[k] = __expf(lg[k] - mx); ssum += e[k]; }
    const float inv = 1.f / ssum;

    if (FINAL) {
      float Sv[K_], s2 = 0.f;
      #pragma unroll
      for (int k = 0; k < K_; ++k) { Sv[k] = e[k]*inv; s2 += Sv[k]; }
      const float invn = 1.f / fmaxf(s2, EPS_);
      #pragma unroll
      for (int k = 0; k < K_; ++k)
        __builtin_nontemporal_store(Sv[k]*invn,
            out + ((size_t)(b*K_ + k))*N_ + pix);        // keep feat in L2
    } else {
      #pragma unroll
      for (int k = 0; k < K_; ++k) {
        float Sv = e[k]*inv;
        dacc[k] += Sv;
        #pragma unroll
        for (int f = 0; f < F_; ++f) cacc[k][f] += Sv * x[f];
      }
    }
  }

  if (!FINAL) {
    // deterministic wave reduce of 56 accumulators, then LDS across 8 waves
    #pragma unroll
    for (int k = 0; k < K_; ++k) {
      #pragma unroll
      for (int o = 16; o > 0; o >>= 1) dacc[k] += __shfl_xor(dacc[k], o, 32);
      #pragma unroll
      for (int f = 0; f < F_; ++f)
        #pragma unroll
        for (int o = 16; o > 0; o >>= 1) cacc[k][f] += __shfl_xor(cacc[k][f], o, 32);
    }
    __shared__ float lsred[8][56];
    if (lid == 0) {
      #pragma unroll
      for (int k = 0; k < K_; ++k) lsred[wid][k] = dacc[k];
      #pragma unroll
      for (int k = 0; k < K_; ++k)
        #pragma unroll
        for (int f = 0; f < F_; ++f) lsred[wid][8 + k*F_ + f] = cacc[k][f];
    }
    __syncthreads();
    if (tid < 56) {
      float s = 0.f;
      #pragma unroll
      for (int w = 0; w < 8; ++w) s += lsred[w][tid];
      partials[((size_t)b*NBLK + blockIdx.x)*56 + tid] = s;
    }
  }
}

// ---------------------------------------------------------------------------
// Centroid update: deterministic reduction of per-block partials.
// partials layout per block: [0..7]=sum S_k, [8 + k*6 + f]=sum S_k*x_f.
// ---------------------------------------------------------------------------
__global__ void update_kernel(const float* __restrict__ partials,
                              float* __restrict__ C) {
  const int t = blockIdx.x * blockDim.x + threadIdx.x;
  if (t >= B_*K_*F_) return;
  const int b  = t / (K_*F_);
  const int kf = t % (K_*F_);
  const int k  = kf / F_;
  const float* pb = partials + (size_t)b*NBLK*56;
  float s = 0.f, d = 0.f;
  for (int blk = 0; blk < NBLK; ++blk) {
    d += pb[blk*56 + k];
    s += pb[blk*56 + 8 + kf];
  }
  C[t] = s / fmaxf(d, EPS_);
}

// ---------------------------------------------------------------------------
extern "C" void kernel_launch(void* const* d_in, const int* in_sizes, int n_in,
                              void* d_out, int out_size, void* d_ws, size_t ws_size,
                              hipStream_t stream) {
  (void)in_sizes; (void)n_in; (void)out_size; (void)ws_size;
  const float* feat = (const float*)d_in[0];    // (B,F,H,W) f32
  const int*   idx  = (const int*)d_in[1];      // (B,K) i32
  float* out = (float*)d_out;                   // (B,K,H,W) f32

  float* ws       = (float*)d_ws;
  float* meanv    = ws;                 // 96
  float* rstdv    = ws + 96;            // 96
  float* C        = ws + 192;           // 768
  float* partials = ws + 960;           // B*NBLK*56 = 57344

  stats_kernel<<<dim3(B_*F_), dim3(256), 0, stream>>>(feat, meanv, rstdv);
  init_centroids<<<dim3(1), dim3(128), 0, stream>>>(feat, idx, meanv, rstdv, C);

  // Reference scan: last S comes from centroids after ITERS-1 updates.
  for (int it = 0; it < ITERS_ - 1; ++it) {
    assign_kernel<false><<<dim3(NBLK, B_), dim3(256), 0, stream>>>(
        feat, meanv, rstdv, C, partials, nullptr);
    update_kernel<<<dim3(3), dim3(256), 0, stream>>>(partials, C);
  }
  assign_kernel<true><<<dim3(NBLK, B_), dim3(256), 0, stream>>>(
      feat, meanv, rstdv, C, nullptr, out);
}